// AssociativeEmbeddingLoss_53068615909932
// MI455X (gfx1250) — compile-verified
//
#include <hip/hip_runtime.h>
#include <math.h>

// Problem constants (fixed by the reference setup)
#define N_IMG 16
#define K_J   17
#define H_    256
#define W_    256
#define M_P   30
#define S_    16          // SCALE_RES
#define MK    (M_P * K_J) // 510

typedef float v2f __attribute__((ext_vector_type(2)));
typedef float v8f __attribute__((ext_vector_type(8)));

// One workgroup per image. 256 threads = 8 wave32s.
// LDS usage ~38 KB (fits easily in 320 KB/WGP).
__global__ __launch_bounds__(256) void ae_loss_img_kernel(
    const float* __restrict__ tags,        // [N, K*H*W] viewed as [N, L, S]
    const int*   __restrict__ joints,      // [N, M, K, 2] int32 (loc, vis)
    const float* __restrict__ box_scales,  // [N, M]
    const float* __restrict__ scale_dist,  // [S]
    float*       __restrict__ ws)          // [N, 3] partials (pull, push, scale)
{
    __shared__ float gath[MK * S_];     // gathered tags (zeros where invisible)
    __shared__ float visf[MK];          // visibility as float
    __shared__ float meanX[32][S_];     // person mean tag, rows 30..31 zero-padded
    __shared__ float cntArr[M_P];
    __shared__ float validf[32];        // rows 30..31 zero
    __shared__ float norms[32];
    __shared__ float pullp[M_P];
    __shared__ float dscale[M_P];
    __shared__ float pushbuf[128];      // per-lane push partials from 4 WMMA waves
    __shared__ float sdist[S_];

    const int n   = blockIdx.x;
    const int tid = threadIdx.x;
    const float* tg = tags   + (size_t)n * (K_J * H_ * W_);
    const int*   jt = joints + (size_t)n * (MK * 2);

    if (tid < S_) sdist[tid] = scale_dist[tid];

    // ---- Phase 1: gather 16 contiguous floats per (m,k); 64B vector loads ----
    for (int idx = tid; idx < MK; idx += 256) {
        const int loc = jt[idx * 2 + 0];
        const int vis = jt[idx * 2 + 1];
        const float vf = (vis > 0) ? 1.0f : 0.0f;
        visf[idx] = vf;
        float4* dst = reinterpret_cast<float4*>(&gath[idx * S_]);
        if (vf > 0.0f) {
            const float4* src = reinterpret_cast<const float4*>(tg + (size_t)loc * S_);
            float4 a0 = src[0], a1 = src[1], a2 = src[2], a3 = src[3];
            dst[0] = a0; dst[1] = a1; dst[2] = a2; dst[3] = a3;
        } else {
            const float4 z = make_float4(0.f, 0.f, 0.f, 0.f);
            dst[0] = z; dst[1] = z; dst[2] = z; dst[3] = z;
        }
    }
    __syncthreads();

    // ---- Phase 2: per-person visible-joint counts ----
    if (tid < M_P) {
        float c = 0.0f;
        for (int k = 0; k < K_J; ++k) c += visf[tid * K_J + k];
        cntArr[tid] = c;
        validf[tid] = (c > 0.0f) ? 1.0f : 0.0f;
    }
    if (tid >= M_P && tid < 32) validf[tid] = 0.0f;
    __syncthreads();

    // ---- Phase 3: mean tag per person (rows 30..31 zero) ----
    for (int idx = tid; idx < 32 * S_; idx += 256) {
        const int m = idx / S_, s = idx % S_;
        float v = 0.0f;
        if (m < M_P) {
            float sum = 0.0f;
            for (int k = 0; k < K_J; ++k) sum += gath[(m * K_J + k) * S_ + s];
            v = sum / fmaxf(cntArr[m], 1.0f);
        }
        meanX[m][s] = v;
    }
    __syncthreads();

    // ---- Phase 4: pull term + scale-cosine term, one thread per person ----
    if (tid < M_P) {
        // pull
        float acc = 0.0f;
        for (int k = 0; k < K_J; ++k) {
            const float vf = visf[tid * K_J + k];
            for (int s = 0; s < S_; ++s) {
                const float d = gath[(tid * K_J + k) * S_ + s] - meanX[tid][s];
                acc += d * d * vf;
            }
        }
        pullp[tid] = acc / (fmaxf(cntArr[tid], 1.0f) * (float)S_);

        // scale: cosine(normalized 1/(gap+1e-10), normalized |mean|)
        const float bs = box_scales[n * M_P + tid];
        float invv[S_], pm[S_];
        float st = 0.0f, sp = 0.0f;
        for (int s = 0; s < S_; ++s) {
            const float gap = fabsf(bs - sdist[s]);
            const float iv  = 1.0f / (gap + 1e-10f);
            invv[s] = iv; st += iv * iv;
            const float p = fabsf(meanX[tid][s]);
            pm[s] = p; sp += p * p;
        }
        const float nt = fmaxf(sqrtf(st), 1e-12f);
        const float np = fmaxf(sqrtf(sp), 1e-12f);
        float dot = 0.0f;
        for (int s = 0; s < S_; ++s) dot += (invv[s] / nt) * (pm[s] / np);
        dscale[tid] = 1.0f - dot;
    }
    // row norms (for push via Gram matrix)
    if (tid < 32) {
        float a = 0.0f;
        for (int s = 0; s < S_; ++s) a += meanX[tid][s] * meanX[tid][s];
        norms[tid] = a;
    }
    __syncthreads();

    // ---- Phase 5: push term via WMMA Gram matrix G = X * X^T, X = [32,16] f32 ----
    // 2x2 tiles of 16x16, waves 0..3 one tile each, K=16 in 4 chained 16x16x4 steps.
    const int wid  = tid >> 5;
    const int lane = tid & 31;
    if (wid < 4) {
        const int i0  = (wid & 1) * 16;     // tile row base (A rows)
        const int j0  = (wid >> 1) * 16;    // tile col base (B cols)
        const int row = lane & 15;
        const int kh  = (lane >> 4) * 2;    // f32 A/B layout: lanes 16-31 hold K+2,K+3
        v8f acc = {0.f, 0.f, 0.f, 0.f, 0.f, 0.f, 0.f, 0.f};
#pragma unroll
        for (int kc = 0; kc < S_; kc += 4) {
            v2f a, b;
            a.x = meanX[i0 + row][kc + kh];
            a.y = meanX[i0 + row][kc + kh + 1];
            b.x = meanX[j0 + row][kc + kh];     // B[k][n] = X[j0+n][k]
            b.y = meanX[j0 + row][kc + kh + 1];
            acc = __builtin_amdgcn_wmma_f32_16x16x4_f32(
                /*neg_a=*/false, a, /*neg_b=*/false, b,
                /*c_mod=*/(short)0, acc, /*reuse_a=*/false, /*reuse_b=*/false);
        }
        // C/D layout: VGPR v holds row v (lanes 0-15) / row v+8 (lanes 16-31), col = lane&15
        float local = 0.0f;
        const int mbase = i0 + ((lane >> 4) ? 8 : 0);
        const int nj    = j0 + (lane & 15);
#pragma unroll
        for (int v = 0; v < 8; ++v) {
            const int mi = mbase + v;
            const float g    = acc[v];
            const float diff = norms[mi] + norms[nj] - 2.0f * g;
            const float mask = validf[mi] * validf[nj] * ((mi != nj) ? 1.0f : 0.0f);
            local += expf(-diff) * mask;
        }
        pushbuf[wid * 32 + lane] = local;
    }
    __syncthreads();

    // ---- Phase 6: deterministic per-image combine ----
    if (tid == 0) {
        float nv = 0.0f;
        for (int m = 0; m < M_P; ++m) nv += validf[m];
        const float safe_n = fmaxf(nv, 1.0f);
        float pull = 0.0f, scl = 0.0f;
        for (int m = 0; m < M_P; ++m) {
            pull += pullp[m]  * validf[m];
            scl  += dscale[m] * validf[m];
        }
        pull /= safe_n;
        scl  /= safe_n;
        float ps = 0.0f;
        for (int i = 0; i < 128; ++i) ps += pushbuf[i];
        const float push = (nv >= 2.0f) ? 0.5f * ps / fmaxf(nv * (nv - 1.0f), 1.0f) : 0.0f;
        ws[n * 3 + 0] = pull;
        ws[n * 3 + 1] = push;
        ws[n * 3 + 2] = scl;
    }
}

// Deterministic final reduction: mean over images, output order (pull, push, scale).
__global__ void ae_loss_reduce_kernel(const float* __restrict__ ws,
                                      float* __restrict__ out)
{
    const int t = threadIdx.x;
    if (t < 3) {
        float s = 0.0f;
        for (int n = 0; n < N_IMG; ++n) s += ws[n * 3 + t];
        out[t] = s / (float)N_IMG;
    }
}

extern "C" void kernel_launch(void* const* d_in, const int* in_sizes, int n_in,
                              void* d_out, int out_size, void* d_ws, size_t ws_size,
                              hipStream_t stream) {
    const float* tags       = (const float*)d_in[0];  // [16,17,256,256] f32
    const int*   joints     = (const int*)  d_in[1];  // [16,30,17,2] int32
    const float* box_scales = (const float*)d_in[2];  // [16,30] f32
    const float* scale_dist = (const float*)d_in[3];  // [16] f32
    float* out = (float*)d_out;                       // 3 f32: pull, push, scale
    float* ws  = (float*)d_ws;                        // 16*3 f32 partials

    ae_loss_img_kernel<<<N_IMG, 256, 0, stream>>>(tags, joints, box_scales,
                                                  scale_dist, ws);
    ae_loss_reduce_kernel<<<1, 32, 0, stream>>>(ws, out);
}